// MultiGIN_11570641895565
// MI455X (gfx1250) — compile-verified
//
#include <hip/hip_runtime.h>

#define N_NODES   50000
#define N_EDGES   800000
#define N_GRAPHS  64
#define D         128
#define ZCOLS     768          // 2 * D * L
#define BN_EPS    1e-5f
#define NEG_SLOPE 0.01f

#define ROWS_PER_BLK 80        // 5 M-tiles of 16; 50000 = 625 * 80
#define MTILES       5
#define LDSW         132       // padded row stride (floats) -> conflict-free
#define A_FLOATS     (ROWS_PER_BLK * LDSW)   // 10560 floats, offset 0
#define W_FLOATS     (D * LDSW)              // 16896 floats
#define POOL_SLICES  8

typedef __attribute__((ext_vector_type(2))) float v2f;
typedef __attribute__((ext_vector_type(8))) float v8f;

// ---------------------------------------------------------------------------
// 16-byte async global->LDS copy (gfx1250 GLOBAL_LOAD_ASYNC_TO_LDS_B128,
// tracked by ASYNCcnt). Generic pointer to __shared__ truncates to the LDS
// byte offset (LDS aperture keeps the offset in addr[31:0]).
// ---------------------------------------------------------------------------
__device__ __forceinline__ void async_copy16(const void* gsrc, void* ldst) {
  unsigned loff = (unsigned)(uintptr_t)ldst;
  asm volatile("global_load_async_to_lds_b128 %0, %1, off"
               :: "v"(loff), "v"(gsrc)
               : "memory");
}
__device__ __forceinline__ void async_wait0() {
  asm volatile("s_wait_asynccnt 0" ::: "memory");
}

// ---------------------------------------------------------------------------
// Edge aggregation: agg[dst] += x[src]   (agg pre-initialized to x via memcpy,
// giving (x + segment_sum) in one buffer). 32 threads per edge, float4 each.
// ---------------------------------------------------------------------------
__global__ __launch_bounds__(256) void aggregate_kernel(
    const float* __restrict__ x, const int* __restrict__ edge,
    float* __restrict__ agg)
{
  unsigned tid = blockIdx.x * 256u + threadIdx.x;
  unsigned e = tid >> 5;                 // edge index (32 feature-groups/edge)
  if (e >= N_EDGES) return;
  unsigned f = (tid & 31u) * 4u;        // feature offset (0..124 step 4)
  int src = edge[e];                    // edge row 0 = source
  int dst = edge[N_EDGES + e];          // edge row 1 = destination
  const float4 v = *(const float4*)(x + (size_t)src * D + f);
  float* p = agg + (size_t)dst * D + f;
  unsafeAtomicAdd(p + 0, v.x);
  unsafeAtomicAdd(p + 1, v.y);
  unsafeAtomicAdd(p + 2, v.z);
  unsafeAtomicAdd(p + 3, v.w);
}

// ---------------------------------------------------------------------------
// FP32 WMMA GEMM + fused epilogue, LDS-staged.
//   out[m,n] = act( sum_k A[m,k] * W[k,n] + bias[n] )
//   mode 0: eval-BN (g,b,m,v) then ReLU;  mode 1: ReLU;  mode 2: LeakyReLU.
// Block = 80 rows x 128 cols. A single __shared__ array is carved manually so
// the A slab is guaranteed at LDS offset 0: all hot-loop ds_load_b64 addresses
// (< 42 KB) fold into the 16-bit DS immediate-offset field. W lives above it
// and is only touched in the one-time B-fragment preload.
// Wave w owns N-tile w; preloads its 64 B-fragment floats once, then runs
// 5 M-tiles x 32 WMMAs from LDS A reads. All 256 threads stay convergent
// (EXEC all-ones for WMMA).
// ---------------------------------------------------------------------------
__global__ __launch_bounds__(256) void gemm_wmma_kernel(
    const float* __restrict__ A, const float* __restrict__ W,
    const float* __restrict__ bias,
    const float* __restrict__ bn_g, const float* __restrict__ bn_b,
    const float* __restrict__ bn_m, const float* __restrict__ bn_v,
    float* __restrict__ out, int mode)
{
  __shared__ float lds[A_FLOATS + W_FLOATS];   // 109824 B total
  float* Alds = lds;                           // offset 0      (42240 B)
  float* Wlds = lds + A_FLOATS;                // offset 42240  (67584 B)

  const int tid  = threadIdx.x;
  const int lane = tid & 31;
  const int wave = tid >> 5;
  const size_t mbase = (size_t)blockIdx.x * ROWS_PER_BLK;

  // ---- stage A slab: 10240 floats, 16B per thread per step, coalesced ----
  #pragma unroll
  for (int it = 0; it < 10; ++it) {
    int idx = it * 1024 + tid * 4;
    int r = idx >> 7, cc = idx & 127;
    async_copy16(A + (mbase + r) * (size_t)D + cc, &Alds[r * LDSW + cc]);
  }
  // ---- stage W: 16384 floats ----
  #pragma unroll
  for (int it = 0; it < 16; ++it) {
    int idx = it * 1024 + tid * 4;
    int r = idx >> 7, cc = idx & 127;
    async_copy16(W + idx, &Wlds[r * LDSW + cc]);
  }
  async_wait0();
  __syncthreads();

  const int half = lane >> 4;                  // 0: lanes 0-15, 1: lanes 16-31
  const int l15  = lane & 15;
  const int n0   = wave * 16;
  const int col  = n0 + l15;                   // this lane's output column

  // ---- preload B fragments for the whole K sweep (reused by all 5 M-tiles)
  // B 4x16 f32 layout: lanes 0-15 hold rows {k,k+1}, lanes 16-31 {k+2,k+3}.
  v2f breg[32];
  #pragma unroll
  for (int s = 0; s < 32; ++s) {
    int kb = 4 * s + (half << 1);
    breg[s].x = Wlds[kb * LDSW + col];
    breg[s].y = Wlds[(kb + 1) * LDSW + col];
  }

  // ---- per-column epilogue constants (uniform branch: EXEC unchanged) ----
  float bias_c = bias[col];
  float sc = 1.0f, sh = 0.0f;
  if (mode == 0) {
    float s0 = bn_g[col] / sqrtf(bn_v[col] + BN_EPS);
    sc = s0;
    sh = bn_b[col] - bn_m[col] * s0;
  }

  #pragma unroll 1
  for (int mt = 0; mt < MTILES; ++mt) {
    v8f c = {};
    // A 16x4 f32 layout: lanes 0-15 hold {k,k+1}, lanes 16-31 {k+2,k+3}
    const float* ar = &Alds[(mt * 16 + l15) * LDSW + (half << 1)];
    #pragma unroll
    for (int s = 0; s < 32; ++s) {
      v2f a = *(const v2f*)(ar + 4 * s);
      c = __builtin_amdgcn_wmma_f32_16x16x4_f32(
          /*neg_a=*/false, a, /*neg_b=*/false, breg[s],
          /*c_mod=*/(short)0, c, /*reuse_a=*/false, /*reuse_b=*/false);
    }
    // C/D layout: VGPR j -> row j + 8*half, column = col
    #pragma unroll
    for (int j = 0; j < 8; ++j) {
      size_t row = mbase + mt * 16 + j + (half << 3);
      float val = c[j] + bias_c;
      if (mode == 0)      { val = val * sc + sh; val = fmaxf(val, 0.0f); }
      else if (mode == 1) { val = fmaxf(val, 0.0f); }
      else                { val = val > 0.0f ? val : NEG_SLOPE * val; }
      out[row * D + col] = val;
    }
  }
}

// ---------------------------------------------------------------------------
// Zero-init of the pooled-feature matrix Z (needed because pool accumulates
// with atomics; workspace contents are undefined at call entry).
// ---------------------------------------------------------------------------
__global__ __launch_bounds__(256) void zero_kernel(float* __restrict__ p, int n)
{
  int i = blockIdx.x * 256 + threadIdx.x;
  if (i < n) p[i] = 0.0f;
}

// ---------------------------------------------------------------------------
// global_add_pool over a *sorted* batch array: grid (graph, slice). Each block
// binary-searches its graph's node range, sums 1/8 of it per feature in
// registers, then atomically accumulates the partial into Z.
// ---------------------------------------------------------------------------
__global__ __launch_bounds__(128) void pool_kernel(
    const float* __restrict__ x, const int* __restrict__ batch,
    float* __restrict__ zseg)
{
  const int g = blockIdx.x;
  const int slice = blockIdx.y;
  const int f = threadIdx.x;
  __shared__ int range[2];
  if (threadIdx.x < 2) {
    int target = g + (int)threadIdx.x;        // lower_bound(g), lower_bound(g+1)
    int lo = 0, hi = N_NODES;
    while (lo < hi) {
      int mid = (lo + hi) >> 1;
      if (batch[mid] < target) lo = mid + 1; else hi = mid;
    }
    range[threadIdx.x] = lo;
  }
  __syncthreads();
  const int lo = range[0], hi = range[1];
  const int per = (hi - lo + POOL_SLICES - 1) / POOL_SLICES;
  const int s0 = lo + slice * per;
  const int s1 = min(s0 + per, hi);
  float s = 0.0f;
  for (int node = s0; node < s1; ++node)
    s += x[(size_t)node * D + f];
  if (s1 > s0)
    unsafeAtomicAdd(&zseg[(size_t)g * ZCOLS + f], s);
}

// ---------------------------------------------------------------------------
// Head: out[g] = relu(Z[g] @ Wf1 + bf1) @ Wf2 + bf2.  64 blocks x 128 threads.
// ---------------------------------------------------------------------------
__global__ __launch_bounds__(128) void head_kernel(
    const float* __restrict__ Z, const float* __restrict__ Wf1,
    const float* __restrict__ bf1, const float* __restrict__ Wf2,
    const float* __restrict__ bf2, float* __restrict__ out)
{
  const int g = blockIdx.x;
  const int n = threadIdx.x;
  float s = bf1[n];
  for (int k = 0; k < ZCOLS; ++k)
    s += Z[(size_t)g * ZCOLS + k] * Wf1[(size_t)k * D + n];
  s = fmaxf(s, 0.0f);
  __shared__ float red[D];
  red[n] = s * Wf2[n];
  __syncthreads();
  for (int off = D / 2; off > 0; off >>= 1) {
    if (n < off) red[n] += red[n + off];
    __syncthreads();
  }
  if (n == 0) out[g] = red[0] + bf2[0];
}

// ---------------------------------------------------------------------------
extern "C" void kernel_launch(void* const* d_in, const int* in_sizes, int n_in,
                              void* d_out, int out_size, void* d_ws, size_t ws_size,
                              hipStream_t stream)
{
  const float* x1   = (const float*)d_in[0];
  const float* x2   = (const float*)d_in[1];
  const float* W1c1 = (const float*)d_in[2];
  const float* b1c1 = (const float*)d_in[3];
  const float* bng  = (const float*)d_in[4];
  const float* bnb  = (const float*)d_in[5];
  const float* bnm  = (const float*)d_in[6];
  const float* bnv  = (const float*)d_in[7];
  const float* W2c1 = (const float*)d_in[8];
  const float* b2c1 = (const float*)d_in[9];
  const float* W1c2 = (const float*)d_in[10];
  const float* b1c2 = (const float*)d_in[11];
  const float* W2c2 = (const float*)d_in[12];
  const float* b2c2 = (const float*)d_in[13];
  const float* Wf1  = (const float*)d_in[14];
  const float* bf1  = (const float*)d_in[15];
  const float* Wf2  = (const float*)d_in[16];
  const float* bf2  = (const float*)d_in[17];
  const int* edge1  = (const int*)d_in[18];
  const int* edge2  = (const int*)d_in[19];
  const int* batch1 = (const int*)d_in[20];
  const int* batch2 = (const int*)d_in[21];

  const size_t NND = (size_t)N_NODES * D;
  float* B0 = (float*)d_ws;                    // 25.6 MB ping
  float* B1 = B0 + NND;                        // 25.6 MB pong
  float* Z  = B1 + NND;                        // [64, 768] pooled features

  const dim3 blk256(256), blk128(128);
  const dim3 aggGrid((N_EDGES * 32 + 255) / 256);   // 100000 blocks
  const dim3 gemmGrid(N_NODES / ROWS_PER_BLK);      // 625 blocks
  const dim3 poolGrid(N_GRAPHS, POOL_SLICES);       // 64 x 8
  float* buf[2] = {B0, B1};

  // Z is accumulated by pool atomics -> zero it first (capture-safe kernel).
  zero_kernel<<<(N_GRAPHS * ZCOLS + 255) / 256, blk256, 0, stream>>>(
      Z, N_GRAPHS * ZCOLS);

  // ---- channel 1: Linear-BN-ReLU-Linear-ReLU (leaky on ReLU output == id) --
  {
    const float* x = x1;
    for (int i = 0; i < 3; ++i) {
      float* agg = buf[i & 1];
      float* hb  = buf[1 - (i & 1)];
      hipMemcpyAsync(agg, x, NND * sizeof(float), hipMemcpyDeviceToDevice, stream);
      aggregate_kernel<<<aggGrid, blk256, 0, stream>>>(x, edge1, agg);
      gemm_wmma_kernel<<<gemmGrid, blk256, 0, stream>>>(
          agg, W1c1 + (size_t)i * D * D, b1c1 + i * D,
          bng + i * D, bnb + i * D, bnm + i * D, bnv + i * D, hb, /*mode=*/0);
      gemm_wmma_kernel<<<gemmGrid, blk256, 0, stream>>>(
          hb, W2c1 + (size_t)i * D * D, b2c1 + i * D,
          nullptr, nullptr, nullptr, nullptr, agg, /*mode=*/1);
      pool_kernel<<<poolGrid, blk128, 0, stream>>>(agg, batch1, Z + i * D);
      x = agg;
    }
  }

  // ---- channel 2: Linear-ReLU-Linear, then LeakyReLU -----------------------
  {
    const float* x = x2;
    for (int i = 0; i < 3; ++i) {
      float* agg = buf[i & 1];
      float* hb  = buf[1 - (i & 1)];
      hipMemcpyAsync(agg, x, NND * sizeof(float), hipMemcpyDeviceToDevice, stream);
      aggregate_kernel<<<aggGrid, blk256, 0, stream>>>(x, edge2, agg);
      gemm_wmma_kernel<<<gemmGrid, blk256, 0, stream>>>(
          agg, W1c2 + (size_t)i * D * D, b1c2 + i * D,
          nullptr, nullptr, nullptr, nullptr, hb, /*mode=*/1);
      gemm_wmma_kernel<<<gemmGrid, blk256, 0, stream>>>(
          hb, W2c2 + (size_t)i * D * D, b2c2 + i * D,
          nullptr, nullptr, nullptr, nullptr, agg, /*mode=*/2);
      pool_kernel<<<poolGrid, blk128, 0, stream>>>(agg, batch2, Z + ZCOLS / 2 + i * D);
      x = agg;
    }
  }

  // ---- head ----------------------------------------------------------------
  head_kernel<<<N_GRAPHS, blk128, 0, stream>>>(Z, Wf1, bf1, Wf2, bf2, (float*)d_out);
}